// InteractionModel_1_52802327937148
// MI455X (gfx1250) — compile-verified
//
#include <hip/hip_runtime.h>
#include <math.h>

// Problem dimensions (fixed by the reference)
#define Bsz 256
#define Lsz 1026
#define Dsz 1280
#define Esz 300
#define Hsz 256
#define SSEG 8   // L-segments for pooling parallelism (deterministic tree reduce)

typedef __attribute__((ext_vector_type(2))) float v2f;
typedef __attribute__((ext_vector_type(4))) float v4f;
typedef __attribute__((ext_vector_type(8))) float v8f;

// ---------------------------------------------------------------------------
// 1) Masked mean pooling, pass 1: per-(batch, L-segment) partial sums.
//    320 threads * float4 = 1280 floats per row l; fully coalesced B128 loads,
//    non-temporal (single-use stream, keep L2 for the reused GEMM weights).
// ---------------------------------------------------------------------------
__global__ void pool_partial_kernel(const float* __restrict__ tok,
                                    const int* __restrict__ lens,
                                    float* __restrict__ partial) {
  const int b = blockIdx.x;
  const int s = blockIdx.y;
  const int t = threadIdx.x;              // 0..319
  const int len = lens[b];
  const int chunk = (Lsz + SSEG - 1) / SSEG;
  int lo = s * chunk; if (lo < 1) lo = 1;
  int hi = (s + 1) * chunk;
  const int lim = len - 1; if (hi > lim) hi = lim;
  const v4f* base = (const v4f*)(tok + (size_t)b * Lsz * Dsz);
  v4f acc = {0.f, 0.f, 0.f, 0.f};
#pragma unroll 4
  for (int l = lo; l < hi; ++l) {
    v4f v = __builtin_nontemporal_load(&base[(size_t)l * (Dsz / 4) + t]);
    acc += v;
  }
  __builtin_nontemporal_store(acc,
      &((v4f*)partial)[((size_t)s * Bsz + b) * (Dsz / 4) + t]);
}

// ---------------------------------------------------------------------------
// 2) Pooling pass 2: deterministic reduce over segments + divide by count.
// ---------------------------------------------------------------------------
__global__ void pool_reduce_kernel(const float* __restrict__ partial,
                                   const int* __restrict__ lens,
                                   float* __restrict__ pooled) {
  const int b = blockIdx.x;
  const int t = threadIdx.x;              // 0..319
  v4f s = {0.f, 0.f, 0.f, 0.f};
#pragma unroll
  for (int ss = 0; ss < SSEG; ++ss) {
    s += ((const v4f*)partial)[((size_t)ss * Bsz + b) * (Dsz / 4) + t];
  }
  int cnt = lens[b] - 2; if (cnt < 1) cnt = 1;
  const float inv = 1.f / (float)cnt;
  ((v4f*)pooled)[(size_t)b * (Dsz / 4) + t] = s * inv;
}

// ---------------------------------------------------------------------------
// 3) Generic f32 GEMM via V_WMMA_F32_16X16X4_F32.
//    C = act(A(MxK) * B(KxN) + bias), one wave per 16x16 tile, K step 4.
//    A layout per lane: row m = lane%16, k = k0 + 2*(lane/16) + {0,1}  (v2f)
//    B layout per lane: col n = lane%16, k = k0 + 2*(lane/16) + {0,1}  (v2f)
//    D layout: acc[r] -> row = 16*mt + r + 8*(lane/16), col = 16*nt + lane%16
//    EDGE=1: ragged N handled by clamped index + zero scale (no EXEC change).
//    EDGE=0: N is a multiple of 16 -> no clamp, no scale VALU in the loop.
// ---------------------------------------------------------------------------
template <int TRANSA, int TRANSB, int EDGE>
__global__ void gemm_wmma_f32(const float* __restrict__ A,
                              const float* __restrict__ Bm,
                              const float* __restrict__ bias,
                              float* __restrict__ C,
                              int M, int N, int K,
                              int lda, int ldb, int ldc, int relu) {
  const int wv = threadIdx.x >> 5;
  const int lane = threadIdx.x & 31;
  const int tilesN = (N + 15) >> 4;
  const int tile = blockIdx.x * (blockDim.x >> 5) + wv;
  if (tile >= (M >> 4) * tilesN) return;     // wave-uniform exit
  const int mt = tile / tilesN;
  const int nt = tile % tilesN;
  const int r   = lane & 15;
  const int hi  = lane >> 4;
  const int m   = mt * 16 + r;
  const int col = nt * 16 + r;
  const int colB = EDGE ? ((col < N) ? col : (N - 1)) : col;
  const float bs = EDGE ? ((col < N) ? 1.f : 0.f) : 1.f;

  v8f acc = {0.f, 0.f, 0.f, 0.f, 0.f, 0.f, 0.f, 0.f};
#pragma unroll 4
  for (int k = 0; k < K; k += 4) {
    const int ka = k + 2 * hi;
    float ax, ay, bx, by;
    if (TRANSA) { ax = A[(size_t)ka * lda + m];       ay = A[(size_t)(ka + 1) * lda + m]; }
    else        { ax = A[(size_t)m * lda + ka];       ay = A[(size_t)m * lda + ka + 1];   }
    if (TRANSB) { bx = Bm[(size_t)colB * ldb + ka];   by = Bm[(size_t)colB * ldb + ka + 1]; }
    else        { bx = Bm[(size_t)ka * ldb + colB];   by = Bm[(size_t)(ka + 1) * ldb + colB]; }
    v2f a = {ax, ay};
    v2f b;
    if (EDGE) { b[0] = bx * bs; b[1] = by * bs; }
    else      { b[0] = bx;      b[1] = by;      }
    acc = __builtin_amdgcn_wmma_f32_16x16x4_f32(false, a, false, b,
                                                (short)0, acc, false, false);
  }

  float bv = 0.f;
  if (bias) bv = bias[colB] * bs;
  if (!EDGE || col < N) {
#pragma unroll
    for (int rr = 0; rr < 8; ++rr) {
      float v = acc[rr] + bv;
      if (relu) v = fmaxf(v, 0.f);
      C[(size_t)(mt * 16 + rr + 8 * hi) * ldc + col] = v;
    }
  }
}

// ---------------------------------------------------------------------------
// 4) Row softmax, in place. One block (256 thr = 8 waves) per row.
// ---------------------------------------------------------------------------
__global__ void softmax_rows_kernel(float* __restrict__ x, int N) {
  const int b = blockIdx.x;
  const int tid = threadIdx.x;
  const int lane = tid & 31;
  const int wv = tid >> 5;
  const int nw = blockDim.x >> 5;
  float* row = x + (size_t)b * N;
  __shared__ float smax[8];
  __shared__ float ssum[8];

  float m = -3.0e38f;
  for (int i = tid; i < N; i += blockDim.x) m = fmaxf(m, row[i]);
  for (int o = 16; o; o >>= 1) m = fmaxf(m, __shfl_xor(m, o, 32));
  if (lane == 0) smax[wv] = m;
  __syncthreads();
  m = smax[0];
  for (int j = 1; j < nw; ++j) m = fmaxf(m, smax[j]);

  float s = 0.f;
  for (int i = tid; i < N; i += blockDim.x) s += expf(row[i] - m);
  for (int o = 16; o; o >>= 1) s += __shfl_xor(s, o, 32);
  if (lane == 0) ssum[wv] = s;
  __syncthreads();
  s = 0.f;
  for (int j = 0; j < nw; ++j) s += ssum[j];
  const float inv = 1.f / s;
  for (int i = tid; i < N; i += blockDim.x) row[i] = expf(row[i] - m) * inv;
}

// ---------------------------------------------------------------------------
// 5) combined = [x1*fx2 + x1, x2*fx1 + x2]
// ---------------------------------------------------------------------------
__global__ void combine_kernel(const float* __restrict__ x1,
                               const float* __restrict__ fx2,
                               const float* __restrict__ x2,
                               const float* __restrict__ fx1,
                               float* __restrict__ comb) {
  const int b = blockIdx.x;
  const int t = threadIdx.x;
  float* cr = comb + (size_t)b * (Dsz + Esz);
  for (int d = t; d < Dsz; d += blockDim.x) {
    const float v = x1[(size_t)b * Dsz + d];
    cr[d] = v * fx2[(size_t)b * Dsz + d] + v;
  }
  for (int e = t; e < Esz; e += blockDim.x) {
    const float v = x2[(size_t)b * Esz + e];
    cr[Dsz + e] = v * fx1[(size_t)b * Esz + e] + v;
  }
}

// ---------------------------------------------------------------------------
// 6) out[b] = dot(h[b,:], W_fc2) + b_fc2   (one wave per row)
// ---------------------------------------------------------------------------
__global__ void fc2_kernel(const float* __restrict__ h,
                           const float* __restrict__ w,
                           const float* __restrict__ b2,
                           float* __restrict__ out) {
  const int wv = threadIdx.x >> 5;
  const int lane = threadIdx.x & 31;
  const int b = blockIdx.x * (blockDim.x >> 5) + wv;
  if (b >= Bsz) return;
  float s = 0.f;
  for (int i = lane; i < Hsz; i += 32) s += h[(size_t)b * Hsz + i] * w[i];
  for (int o = 16; o; o >>= 1) s += __shfl_xor(s, o, 32);
  if (lane == 0) out[b] = s + b2[0];
}

// ---------------------------------------------------------------------------
extern "C" void kernel_launch(void* const* d_in, const int* in_sizes, int n_in,
                              void* d_out, int out_size, void* d_ws, size_t ws_size,
                              hipStream_t stream) {
  (void)in_sizes; (void)n_in; (void)out_size; (void)ws_size;
  const float* tok  = (const float*)d_in[0];
  const int*   lens = (const int*)  d_in[1];
  const float* x2   = (const float*)d_in[2];
  const float* Wp   = (const float*)d_in[3];
  const float* bp   = (const float*)d_in[4];
  const float* Wa   = (const float*)d_in[5];
  const float* ba   = (const float*)d_in[6];
  const float* W1   = (const float*)d_in[7];
  const float* b1   = (const float*)d_in[8];
  const float* W2   = (const float*)d_in[9];
  const float* b2   = (const float*)d_in[10];
  float* out = (float*)d_out;
  float* ws  = (float*)d_ws;

  // Workspace layout. "partial" (SSEG*B*D floats) is dead after pool_reduce,
  // so the attention-stage buffers are aliased into it (they total < its size).
  float* partial = ws;                                       // SSEG*B*D
  float* y       = ws;                                       // B*E
  float* scores  = y      + (size_t)Bsz * Esz;               // B*B  (-> attn in place)
  float* t1      = scores + (size_t)Bsz * Bsz;               // B*E  (-> fused_x1 in place)
  float* t2      = t1     + (size_t)Bsz * Esz;               // B*D  (-> fused_x2 in place)
  float* comb    = t2     + (size_t)Bsz * Dsz;               // B*(D+E)
  float* hbuf    = comb   + (size_t)Bsz * (Dsz + Esz);       // B*H
  float* pooled  = ws + (size_t)SSEG * Bsz * Dsz;            // B*D (outside partial)
  float* x1buf   = pooled + (size_t)Bsz * Dsz;               // B*D (outside partial)

  // --- pooling (bandwidth-dominant pass: ~1.34 GB streamed once) ---
  pool_partial_kernel<<<dim3(Bsz, SSEG), Dsz / 4, 0, stream>>>(tok, lens, partial);
  pool_reduce_kernel<<<Bsz, Dsz / 4, 0, stream>>>(partial, lens, pooled);

  const int wavesPerBlock = 8;           // 256 threads
  auto blocksFor = [](int M, int N) { return ((M / 16) * ((N + 15) / 16) + 7) / 8; };

  // x1 = relu(pooled @ W_prot + b_prot)              [256,1280]
  gemm_wmma_f32<0, 0, 0><<<blocksFor(Bsz, Dsz), 32 * wavesPerBlock, 0, stream>>>(
      pooled, Wp, bp, x1buf, Bsz, Dsz, Dsz, Dsz, Dsz, Dsz, 1);

  // y = x1 @ W_attn + b_attn                         [256,300] (ragged N)
  gemm_wmma_f32<0, 0, 1><<<blocksFor(Bsz, Esz), 32 * wavesPerBlock, 0, stream>>>(
      x1buf, Wa, ba, y, Bsz, Esz, Dsz, Dsz, Esz, Esz, 0);

  // scores = y @ x2^T                                [256,256]
  gemm_wmma_f32<0, 1, 0><<<blocksFor(Bsz, Bsz), 32 * wavesPerBlock, 0, stream>>>(
      y, x2, nullptr, scores, Bsz, Bsz, Esz, Esz, Esz, Bsz, 0);

  // attn = softmax(scores, axis=1)  (in place)
  softmax_rows_kernel<<<Bsz, 256, 0, stream>>>(scores, Bsz);

  // t1 = attn @ x2                                   [256,300] (ragged N)
  gemm_wmma_f32<0, 0, 1><<<blocksFor(Bsz, Esz), 32 * wavesPerBlock, 0, stream>>>(
      scores, x2, nullptr, t1, Bsz, Esz, Bsz, Bsz, Esz, Esz, 0);
  // fused_x1 = softmax(t1, axis=1)
  softmax_rows_kernel<<<Bsz, 256, 0, stream>>>(t1, Esz);

  // t2 = attn^T @ x1                                 [256,1280]
  gemm_wmma_f32<1, 0, 0><<<blocksFor(Bsz, Dsz), 32 * wavesPerBlock, 0, stream>>>(
      scores, x1buf, nullptr, t2, Bsz, Dsz, Bsz, Bsz, Dsz, Dsz, 0);
  // fused_x2 = softmax(t2, axis=1)
  softmax_rows_kernel<<<Bsz, 256, 0, stream>>>(t2, Dsz);

  // combined = [x1*fused_x2 + x1, x2*fused_x1 + x2]  [256,1580]
  combine_kernel<<<Bsz, 256, 0, stream>>>(x1buf, t2, x2, t1, comb);

  // h = relu(combined @ W_fc1 + b_fc1)               [256,256], K=1580
  gemm_wmma_f32<0, 0, 0><<<blocksFor(Bsz, Hsz), 32 * wavesPerBlock, 0, stream>>>(
      comb, W1, b1, hbuf, Bsz, Hsz, Dsz + Esz, Dsz + Esz, Hsz, Hsz, 1);

  // out = h @ W_fc2 + b_fc2                          [256,1]
  fc2_kernel<<<Bsz / wavesPerBlock, 32 * wavesPerBlock, 0, stream>>>(hbuf, W2, b2, out);
}